// BilinearTensorProduct_56006373540409
// MI455X (gfx1250) — compile-verified
//
#include <hip/hip_runtime.h>
#include <math.h>

// MI455X / gfx1250, wave32. Bilinear tensor product:
//   out = tanh( concat(V1,V2)@W  +  einsum('bd,kde,be->bk',V1,T,V2) + sum(b) )
// Bilinear term as GEMM  P @ T_flat^T  with P[b, d*256+e] = V1[b,d]*V2[b,e] built
// on the fly (v_pk_mul_f32); accumulated with V_WMMA_F32_16X16X4_F32 (exact f32).
// T is streamed L2->LDS with GLOBAL_LOAD_ASYNC_TO_LDS_B128 (ASYNCcnt), double-buffered.

typedef float v2f __attribute__((ext_vector_type(2)));
typedef float v4f __attribute__((ext_vector_type(4)));
typedef float v8f __attribute__((ext_vector_type(8)));
typedef int   v4i __attribute__((ext_vector_type(4)));

#define BDIM   8192
#define DDIM   256
#define KOUT   64
#define BT     64            // batch rows per workgroup
#define TROW   132           // dwords per staged T row (128 e + 4 pad -> bank stride 4)
#define TBUFD  (64 * TROW)   // dwords per T double-buffer half (64 k rows)

// LDS layout (floats):
//  sV1T : [256 d][64 m]                                      16384
//  sV2S : [64 g][4 mt][32 lane]{2}  (WMMA A-fragment pairs)  16384
//  sT0/sT1 : [64 k][132]                                     2*8448
//  sSum : 1
#define LDS_FLOATS (16384 + 16384 + 2 * TBUFD + 1)
#define LDS_BYTES  (LDS_FLOATS * 4)

#if __has_builtin(__builtin_amdgcn_global_load_async_to_lds_b128)
#define HAVE_ASYNC_LDS 1
#endif

__device__ __forceinline__ void wait_asynccnt0() {
#if __has_builtin(__builtin_amdgcn_s_wait_asynccnt)
    __builtin_amdgcn_s_wait_asynccnt(0);
#else
    asm volatile("s_wait_asynccnt 0x0" ::: "memory");
#endif
}

__global__ __launch_bounds__(256)
void btp_gfx1250_kernel(const float* __restrict__ V1, const float* __restrict__ V2,
                        const float* __restrict__ T,  const float* __restrict__ W,
                        const float* __restrict__ bvec, float* __restrict__ out)
{
    extern __shared__ float smem[];
    float* sV1T = smem;                         // transposed V1 tile
    float* sV2S = smem + 16384;                 // lane-swizzled V2 pairs
    float* sT0  = sV2S + 16384;
    float* sT1  = sT0 + TBUFD;
    float* sSum = sT1 + TBUFD;

    const int tid  = threadIdx.x;
    const int lane = tid & 31;
    const int wid  = tid >> 5;     // 0..7
    const int hv   = wid >> 2;     // batch half: rows hv*32 .. hv*32+31
    const int kt   = wid & 3;      // k-tile: cols kt*16 .. kt*16+15
    const int n    = lane & 15;
    const int hi   = lane >> 4;    // 0: K=0,1 ; 1: K=2,3 (WMMA A/B fragment halves)
    const int sel  = hi * 2;
    const int b0   = blockIdx.x * BT;

    if (tid == 0) sSum[0] = 0.0f;

    // ---- Stage V1 (transposed) and V2 (WMMA-swizzled pairs) into LDS ----
    for (int s = tid; s < BT * (DDIM / 4); s += 256) {   // 4096 slots, exact
        const int m = s >> 6;        // 0..63 batch row in tile
        const int g = s & 63;        // group of 4 along d/e
        v4f a = *(const v4f*)(V1 + (size_t)(b0 + m) * DDIM + g * 4);
        sV1T[(g * 4 + 0) * 64 + m] = a.x;
        sV1T[(g * 4 + 1) * 64 + m] = a.y;
        sV1T[(g * 4 + 2) * 64 + m] = a.z;
        sV1T[(g * 4 + 3) * 64 + m] = a.w;
        v4f c = *(const v4f*)(V2 + (size_t)(b0 + m) * DDIM + g * 4);
        const int mt2 = m >> 4, ml = m & 15;
        float* dst = sV2S + ((g * 4 + mt2) * 32) * 2;
        dst[ml * 2 + 0] = c.x;  dst[ml * 2 + 1] = c.y;               // lane ml   : K=0,1
        dst[(16 + ml) * 2 + 0] = c.z;  dst[(16 + ml) * 2 + 1] = c.w; // lane 16+ml: K=2,3
    }
    __syncthreads();
    atomicAdd(sSum, bvec[tid]);      // ds_add_f32; 256 threads cover all of b

    v8f acc0 = {0.f,0.f,0.f,0.f,0.f,0.f,0.f,0.f};
    v8f acc1 = acc0;

    // ---- T chunk staging: async DMA global->LDS, no VGPR data ----
    auto load_chunk = [&](int chunk, float* dstT) {
        const int d  = chunk >> 1;
        const int eh = chunk & 1;
        const float* src = T + (size_t)d * DDIM + (size_t)eh * 128;
        #pragma unroll
        for (int i = 0; i < 8; ++i) {                    // 2048 b128 slots / 256 thr
            const int s   = tid + i * 256;
            const int row = s >> 5;
            const int c4  = (s & 31) * 4;
            const float* g  = src  + (size_t)row * (DDIM * DDIM) + c4;
            float*       l  = dstT + row * TROW + c4;
#ifdef HAVE_ASYNC_LDS
            __builtin_amdgcn_global_load_async_to_lds_b128((v4i*)g, (v4i*)l, 0, 0);
#else
            *(v4f*)l = *(const v4f*)g;
#endif
        }
    };

    // ---- Feedforward path folded into the same accumulators ----
    const int kbase = kt * 16;
    #pragma unroll 4
    for (int g = 0; g < 64; ++g) {               // j = 4g .. from V1
        const int j = 4 * g + sel;
        v2f bm = { W[(size_t)j * KOUT + kbase + n],
                   W[(size_t)(j + 1) * KOUT + kbase + n] };
        v2f a0 = { sV1T[j * 64 + hv * 32 + n],      sV1T[(j + 1) * 64 + hv * 32 + n] };
        v2f a1 = { sV1T[j * 64 + hv * 32 + 16 + n], sV1T[(j + 1) * 64 + hv * 32 + 16 + n] };
        acc0 = __builtin_amdgcn_wmma_f32_16x16x4_f32(false, a0, false, bm, (short)0, acc0, false, false);
        acc1 = __builtin_amdgcn_wmma_f32_16x16x4_f32(false, a1, false, bm, (short)0, acc1, false, false);
    }
    #pragma unroll 4
    for (int g = 0; g < 64; ++g) {               // j = 256+4g .. from V2 (swizzled pairs ARE the A-fragment)
        const int j = 256 + 4 * g + sel;
        v2f bm = { W[(size_t)j * KOUT + kbase + n],
                   W[(size_t)(j + 1) * KOUT + kbase + n] };
        v2f a0 = *(const v2f*)(sV2S + ((g * 4 + hv * 2 + 0) * 32 + lane) * 2);
        v2f a1 = *(const v2f*)(sV2S + ((g * 4 + hv * 2 + 1) * 32 + lane) * 2);
        acc0 = __builtin_amdgcn_wmma_f32_16x16x4_f32(false, a0, false, bm, (short)0, acc0, false, false);
        acc1 = __builtin_amdgcn_wmma_f32_16x16x4_f32(false, a1, false, bm, (short)0, acc1, false, false);
    }

    // ---- Bilinear path: stream T through LDS (double-buffered) ----
    load_chunk(0, sT0);
    wait_asynccnt0();
    __syncthreads();

    for (int chunk = 0; chunk < 512; ++chunk) {          // (d, e-half)
        float* cur = (chunk & 1) ? sT1 : sT0;
        float* nxt = (chunk & 1) ? sT0 : sT1;
        if (chunk + 1 < 512) load_chunk(chunk + 1, nxt);

        const int d = chunk >> 1;
        const float v1a = sV1T[d * 64 + hv * 32 + n];        // broadcast read
        const float v1b = sV1T[d * 64 + hv * 32 + 16 + n];
        const float* trow = cur + (kt * 16 + n) * TROW + sel;
        const float* v2base = sV2S + (((chunk & 1) * 32) * 4 + hv * 2) * 32 * 2 + lane * 2;

        #pragma unroll
        for (int gl = 0; gl < 32; ++gl) {
            v2f bm = *(const v2f*)(trow + gl * 4);                // T[k_n, d, e0+sel..+1]
            v2f u0 = *(const v2f*)(v2base + (gl * 4 + 0) * 64);   // V2 pair, m-tile 0
            v2f u1 = *(const v2f*)(v2base + (gl * 4 + 1) * 64);   // V2 pair, m-tile 1
            v2f a0 = u0 * v1a;                                    // v_pk_mul_f32
            v2f a1 = u1 * v1b;
            acc0 = __builtin_amdgcn_wmma_f32_16x16x4_f32(false, a0, false, bm, (short)0, acc0, false, false);
            acc1 = __builtin_amdgcn_wmma_f32_16x16x4_f32(false, a1, false, bm, (short)0, acc1, false, false);
        }
        wait_asynccnt0();        // next chunk's DMA landed in LDS
        __syncthreads();
    }

    // ---- Epilogue: + sum(b), tanh, store (16 coalesced b32 per lane) ----
    const float sb = sSum[0];
    const int col = kt * 16 + n;
    #pragma unroll
    for (int r = 0; r < 8; ++r) {
        const int m = b0 + hv * 32 + r + hi * 8;
        out[(size_t)m * KOUT + col]        = tanhf(acc0[r] + sb);
        out[(size_t)(m + 16) * KOUT + col] = tanhf(acc1[r] + sb);
    }
}

extern "C" void kernel_launch(void* const* d_in, const int* in_sizes, int n_in,
                              void* d_out, int out_size, void* d_ws, size_t ws_size,
                              hipStream_t stream) {
    (void)in_sizes; (void)n_in; (void)out_size; (void)d_ws; (void)ws_size;
    const float* V1 = (const float*)d_in[0];
    const float* V2 = (const float*)d_in[1];
    const float* T  = (const float*)d_in[2];
    const float* W  = (const float*)d_in[3];
    const float* bv = (const float*)d_in[4];
    float* out = (float*)d_out;

    (void)hipFuncSetAttribute((const void*)btp_gfx1250_kernel,
                              hipFuncAttributeMaxDynamicSharedMemorySize, LDS_BYTES);
    btp_gfx1250_kernel<<<dim3(BDIM / BT), dim3(256), LDS_BYTES, stream>>>(V1, V2, T, W, bv, out);
}